// CAL_LOSS_PSAB_13142599926297
// MI455X (gfx1250) — compile-verified
//
#include <hip/hip_runtime.h>
#include <hip/hip_bf16.h>
#include <math.h>

#define VOCAB  50257
#define DDIM   256
#define NC     4
#define NTOK   4096            // B*W = 4*1024
#define TEMPF  1.0f

typedef __attribute__((ext_vector_type(2))) float v2f;
typedef __attribute__((ext_vector_type(4))) float v4f;
typedef __attribute__((ext_vector_type(8))) float v8f;

// ---------------------------------------------------------------------------
// Stateless hash -> Gumbel(0,1) noise. Deterministic per (row, vocab id).
// ---------------------------------------------------------------------------
__device__ __forceinline__ float gumbel_from_key(unsigned key) {
    unsigned h = key * 0x9E3779B9u;
    h ^= h >> 16; h *= 0x85EBCA6Bu;
    h ^= h >> 13; h *= 0xC2B2AE35u;
    h ^= h >> 16;
    // u in (0,1): 24-bit mantissa + half-ulp offset, never 0 or 1
    float u = (float)(h >> 8) * (1.0f / 16777216.0f) + (1.0f / 33554432.0f);
    return -__logf(-__logf(u));
}

// ---------------------------------------------------------------------------
// Kernel 1: per token row, top-4 of (logit + gumbel) with label excluded,
// then place label at the random answer slot.  One block per token.
// Bandwidth-bound: streams 823 MB of logits once (NT 128-bit loads).
// ---------------------------------------------------------------------------
__global__ void __launch_bounds__(256)
topk_sample_kernel(const float* __restrict__ logits,
                   const int*   __restrict__ labels,
                   int*         __restrict__ mctask,   // [NTOK][NC]
                   int*         __restrict__ answer)   // [NTOK]
{
    const int row   = blockIdx.x;
    const int t     = threadIdx.x;
    const int label = labels[row];

    const size_t rowstart = (size_t)row * VOCAB;
    const float* base     = logits + rowstart;
    const unsigned rbase  = (unsigned)rowstart;

    float tv0 = -INFINITY, tv1 = -INFINITY, tv2 = -INFINITY, tv3 = -INFINITY;
    int   ti0 = 0, ti1 = 0, ti2 = 0, ti3 = 0;

    auto consider = [&](int v, float x) {
        if (v == label) return;
        float s = x + gumbel_from_key(rbase + (unsigned)v);
        if (s > tv3) {
            if (s > tv0)      { tv3=tv2; ti3=ti2; tv2=tv1; ti2=ti1; tv1=tv0; ti1=ti0; tv0=s; ti0=v; }
            else if (s > tv1) { tv3=tv2; ti3=ti2; tv2=tv1; ti2=ti1; tv1=s;  ti1=v; }
            else if (s > tv2) { tv3=tv2; ti3=ti2; tv2=s;  ti2=v; }
            else              { tv3=s;  ti3=v; }
        }
    };

    // Align to 16B: scalar prologue, then NT float4 stream, then scalar tail.
    const int pro = (int)((4 - (rowstart & 3)) & 3);
    if (t < pro) consider(t, base[t]);
    const int n4 = (VOCAB - pro) >> 2;
    const v4f* b4 = reinterpret_cast<const v4f*>(base + pro);
    for (int i = t; i < n4; i += 256) {
        v4f x = __builtin_nontemporal_load(b4 + i);      // global_load_b128, TH=NT
        const int v = pro + (i << 2);
        consider(v + 0, x.x); consider(v + 1, x.y);
        consider(v + 2, x.z); consider(v + 3, x.w);
    }
    for (int v = pro + (n4 << 2) + t; v < VOCAB; v += 256) consider(v, base[v]);

    // Block-wide top-4 via 4 argmax passes over the 1024 per-lane candidates.
    __shared__ float sval[1024];
    __shared__ int   sidx[1024];
    sval[t*4+0]=tv0; sidx[t*4+0]=ti0;
    sval[t*4+1]=tv1; sidx[t*4+1]=ti1;
    sval[t*4+2]=tv2; sidx[t*4+2]=ti2;
    sval[t*4+3]=tv3; sidx[t*4+3]=ti3;
    __syncthreads();

    __shared__ float rv[256];
    __shared__ int   ri[256];
    __shared__ int   rs[256];
    __shared__ int   topidx[NC];

    for (int pass = 0; pass < NC; ++pass) {
        float best = -INFINITY; int bi = 0, bs = -1;
        #pragma unroll
        for (int j = t; j < 1024; j += 256) {
            float vv = sval[j];
            if (vv > best) { best = vv; bi = sidx[j]; bs = j; }
        }
        rv[t] = best; ri[t] = bi; rs[t] = bs;
        __syncthreads();
        for (int off = 128; off > 0; off >>= 1) {
            if (t < off && rv[t + off] > rv[t]) {
                rv[t] = rv[t + off]; ri[t] = ri[t + off]; rs[t] = rs[t + off];
            }
            __syncthreads();
        }
        if (t == 0) {
            topidx[pass] = ri[0];
            if (rs[0] >= 0) sval[rs[0]] = -INFINITY;   // knock out winner
        }
        __syncthreads();
    }

    if (t == 0) {
        unsigned h = ((unsigned)row ^ 0xA5A5F00Du) * 0x9E3779B9u;
        h ^= h >> 16; h *= 0x85EBCA6Bu; h ^= h >> 13; h *= 0xC2B2AE35u; h ^= h >> 16;
        const int ans = (int)(h & 3u);
        answer[row] = ans;
        #pragma unroll
        for (int c = 0; c < NC; ++c)
            mctask[row * NC + c] = (c == ans) ? label : topidx[c];
    }
}

// ---------------------------------------------------------------------------
// Kernel 2: 16 tokens per wave.  For each choice c, WMMA
// D = Emb_rows(16 x 256) * Datax^T(256 x 16) accumulated f32 over K in steps
// of 4 via V_WMMA_F32_16X16X4_F32; only the diagonal D[t][t] is used.
// Then 4-way log-softmax CE, masked, atomically accumulated.
// ---------------------------------------------------------------------------
__global__ void __launch_bounds__(32)
mc_wmma_ce_kernel(const float* __restrict__ datax,
                  const float* __restrict__ pt_emb,
                  const float* __restrict__ pt_emb_bias,
                  const float* __restrict__ input_mask,
                  const int*   __restrict__ mctask,
                  const int*   __restrict__ answer,
                  float*       __restrict__ acc)      // [0]=sum ce*w  [1]=sum w
{
    __shared__ __align__(16) float xs[16][DDIM + 4];  // +4 pad: stride 260 kills bank conflicts
    __shared__ float cds[NC][32][8];
    __shared__ int   rows_s[16][NC];

    const int lane = threadIdx.x;
    const int tok0 = blockIdx.x * 16;

    // Stage 16 tokens x 256 floats of datax into LDS (coalesced float4).
    const v4f* dx4 = reinterpret_cast<const v4f*>(datax + (size_t)tok0 * DDIM);
    for (int i = lane; i < 1024; i += 32) {
        const int r = i >> 6, c4 = i & 63;
        *reinterpret_cast<v4f*>(&xs[r][c4 << 2]) = dx4[i];
    }
    if (lane < 16) {
        #pragma unroll
        for (int c = 0; c < NC; ++c)
            rows_s[lane][c] = mctask[(tok0 + lane) * NC + c];
    }
    __syncthreads();

    const int m  = lane & 15;          // A row (emb of token m) and B col (datax of token m)
    const int kh = (lane >> 4) << 1;   // K sub-offset 0 or 2 per ISA A/B layout

    const float* a0 = pt_emb + (size_t)rows_s[m][0] * DDIM;
    const float* a1 = pt_emb + (size_t)rows_s[m][1] * DDIM;
    const float* a2 = pt_emb + (size_t)rows_s[m][2] * DDIM;
    const float* a3 = pt_emb + (size_t)rows_s[m][3] * DDIM;

    v8f c0 = {}; v8f c1 = {}; v8f c2 = {}; v8f c3 = {};

    for (int kb = 0; kb < DDIM; kb += 4) {
        const v2f b = *reinterpret_cast<const v2f*>(&xs[m][kb + kh]);
        v2f a;
        a = *reinterpret_cast<const v2f*>(a0 + kb + kh);
        c0 = __builtin_amdgcn_wmma_f32_16x16x4_f32(false, a, false, b, (short)0, c0, false, false);
        a = *reinterpret_cast<const v2f*>(a1 + kb + kh);
        c1 = __builtin_amdgcn_wmma_f32_16x16x4_f32(false, a, false, b, (short)0, c1, false, false);
        a = *reinterpret_cast<const v2f*>(a2 + kb + kh);
        c2 = __builtin_amdgcn_wmma_f32_16x16x4_f32(false, a, false, b, (short)0, c2, false, false);
        a = *reinterpret_cast<const v2f*>(a3 + kb + kh);
        c3 = __builtin_amdgcn_wmma_f32_16x16x4_f32(false, a, false, b, (short)0, c3, false, false);
    }

    // Spill D matrices to LDS so each token-lane can read its diagonal.
    #pragma unroll
    for (int j = 0; j < 8; ++j) {
        cds[0][lane][j] = c0[j];
        cds[1][lane][j] = c1[j];
        cds[2][lane][j] = c2[j];
        cds[3][lane][j] = c3[j];
    }
    __syncthreads();

    float ce_w = 0.0f, wsum = 0.0f;
    if (lane < 16) {
        const int tok = tok0 + lane;
        float o0, o1, o2, o3;
        {
            // D[t][t]: t<8 -> (lane=t, vgpr=t); t>=8 -> (lane=t+16, vgpr=t-8)
            const int sl = (lane < 8) ? lane : lane + 16;
            const int sj = (lane < 8) ? lane : lane - 8;
            o0 = TEMPF * (cds[0][sl][sj] + pt_emb_bias[rows_s[lane][0]]);
            o1 = TEMPF * (cds[1][sl][sj] + pt_emb_bias[rows_s[lane][1]]);
            o2 = TEMPF * (cds[2][sl][sj] + pt_emb_bias[rows_s[lane][2]]);
            o3 = TEMPF * (cds[3][sl][sj] + pt_emb_bias[rows_s[lane][3]]);
        }
        const float mx  = fmaxf(fmaxf(o0, o1), fmaxf(o2, o3));
        const float s   = __expf(o0 - mx) + __expf(o1 - mx) + __expf(o2 - mx) + __expf(o3 - mx);
        const float lse = mx + __logf(s);
        const int   ans = answer[tok];
        const float oa  = (ans == 0) ? o0 : (ans == 1) ? o1 : (ans == 2) ? o2 : o3;
        const float wm  = 1.0f - input_mask[tok];
        ce_w = (lse - oa) * wm;
        wsum = wm;
    }
    // wave32 reduction
    for (int off = 16; off > 0; off >>= 1) {
        ce_w += __shfl_down(ce_w, off, 32);
        wsum += __shfl_down(wsum, off, 32);
    }
    if (lane == 0) {
        atomicAdd(&acc[0], ce_w);
        atomicAdd(&acc[1], wsum);
    }
}

__global__ void init_acc_kernel(float* acc) {
    if (threadIdx.x < 2) acc[threadIdx.x] = 0.0f;
}

__global__ void finalize_kernel(const float* acc, float* out) {
    out[0] = acc[0] / acc[1];
}

// ---------------------------------------------------------------------------
extern "C" void kernel_launch(void* const* d_in, const int* in_sizes, int n_in,
                              void* d_out, int out_size, void* d_ws, size_t ws_size,
                              hipStream_t stream) {
    const float* datax       = (const float*)d_in[0];   // [4,1024,256]
    const float* logits      = (const float*)d_in[1];   // [4,1024,50257]
    const int*   labels      = (const int*)  d_in[2];   // [4,1024]
    const float* pt_emb      = (const float*)d_in[3];   // [50257,256]
    const float* pt_emb_bias = (const float*)d_in[4];   // [50257]
    const float* input_mask  = (const float*)d_in[5];   // [4,1024]
    float* out = (float*)d_out;

    char* ws = (char*)d_ws;
    int*   mctask = (int*)ws;                                              // 64 KB
    int*   answer = (int*)(ws + (size_t)NTOK * NC * sizeof(int));          // 16 KB
    float* acc    = (float*)(ws + (size_t)NTOK * NC * sizeof(int)
                                + (size_t)NTOK * sizeof(int));             // 8 B

    init_acc_kernel<<<1, 32, 0, stream>>>(acc);
    topk_sample_kernel<<<NTOK, 256, 0, stream>>>(logits, labels, mctask, answer);
    mc_wmma_ce_kernel<<<NTOK / 16, 32, 0, stream>>>(datax, pt_emb, pt_emb_bias,
                                                    input_mask, mctask, answer, acc);
    finalize_kernel<<<1, 1, 0, stream>>>(acc, out);
}